// IMP_GCN_47382079210023
// MI455X (gfx1250) — compile-verified
//
#include <hip/hip_runtime.h>

typedef __attribute__((ext_vector_type(2))) float v2f;
typedef __attribute__((ext_vector_type(8))) float v8f;

#define DIM 64

// ---------------- init: ego = concat(user, up); out = 0.8 * ego (layer-0 term) ----
__global__ void k_init(const float* __restrict__ user, const float* __restrict__ up,
                       float* __restrict__ ego, float* __restrict__ out,
                       int nuElems, long long nd) {
    long long i = (long long)blockIdx.x * blockDim.x + threadIdx.x;
    if (i >= nd) return;
    float v = (i < nuElems) ? user[i] : up[i - nuElems];
    ego[i] = v;
    out[i] = 0.8f * v;   // GROUPS * LAYER_W = 4 * 0.2
}

__global__ void k_zero_i32(int* __restrict__ p, int n) {
    int i = blockIdx.x * blockDim.x + threadIdx.x;
    if (i < n) p[i] = 0;
}

// ---------------- CSR build -------------------------------------------------------
__global__ void k_count(const int* __restrict__ rows, int* __restrict__ counts, int E) {
    int e = blockIdx.x * blockDim.x + threadIdx.x;
    if (e < E) atomicAdd(&counts[rows[e]], 1);
}

__global__ void k_scan_block(int* __restrict__ data, int n, int* __restrict__ partials) {
    __shared__ int sh[256];
    int g = blockIdx.x * 256 + threadIdx.x;
    int v = (g < n) ? data[g] : 0;
    sh[threadIdx.x] = v;
    __syncthreads();
    for (int off = 1; off < 256; off <<= 1) {
        int t = (threadIdx.x >= off) ? sh[threadIdx.x - off] : 0;
        __syncthreads();
        sh[threadIdx.x] += t;
        __syncthreads();
    }
    if (g < n) data[g] = sh[threadIdx.x] - v;        // exclusive within block
    if (threadIdx.x == 255) partials[blockIdx.x] = sh[255];
}

__global__ void k_scan_partials(int* __restrict__ partials, int nb) {
    __shared__ int sh[1024];
    int v = (threadIdx.x < nb) ? partials[threadIdx.x] : 0;
    sh[threadIdx.x] = v;
    __syncthreads();
    for (int off = 1; off < 1024; off <<= 1) {
        int t = (threadIdx.x >= off) ? sh[threadIdx.x - off] : 0;
        __syncthreads();
        sh[threadIdx.x] += t;
        __syncthreads();
    }
    if (threadIdx.x < nb) partials[threadIdx.x] = sh[threadIdx.x] - v;  // exclusive
}

__global__ void k_scan_add(int* __restrict__ data, int n, const int* __restrict__ partials,
                           int* __restrict__ cursor, int N) {
    int g = blockIdx.x * 256 + threadIdx.x;
    if (g >= n) return;
    int v = data[g] + partials[blockIdx.x];
    data[g] = v;
    if (g < N) cursor[g] = v;
}

// scatter: edge payload packed as int2 {col, val_bits} -> one b64 load per edge later
__global__ void k_scatter(const int* __restrict__ rows, const int* __restrict__ cols,
                          const float* __restrict__ vals, int* __restrict__ cursor,
                          int2* __restrict__ csr_cv, int E) {
    int e = blockIdx.x * blockDim.x + threadIdx.x;
    if (e >= E) return;
    int p = atomicAdd(&cursor[rows[e]], 1);
    int2 cv;
    cv.x = cols[e];
    cv.y = __float_as_int(vals[e]);
    csr_cv[p] = cv;
}

// after group routing: repack {(col<<4)|mask[col], val_bits} (col < 2^18 so <<4 is safe)
__global__ void k_pack_mask(const int2* __restrict__ csr_cv, const int* __restrict__ gmask,
                            int2* __restrict__ csr_cvm, int E) {
    int e = blockIdx.x * blockDim.x + threadIdx.x;
    if (e >= E) return;
    int2 cv = csr_cv[e];
    int2 o;
    o.x = (cv.x << 4) | gmask[cv.x];
    o.y = cv.y;
    csr_cvm[e] = o;
}

// ---------------- SpMM: h = ego + A*ego  (one wave32 per row, lane owns 2 dims) ---
__global__ void k_spmm_h(const int* __restrict__ row_off, const int2* __restrict__ csr_cv,
                         const float* __restrict__ ego, float* __restrict__ h, int N) {
    int gid  = blockIdx.x * blockDim.x + threadIdx.x;
    int r    = gid >> 5;
    int lane = threadIdx.x & 31;
    if (r >= N) return;
    int s = row_off[r], e = row_off[r + 1];
    float a0 = 0.f, a1 = 0.f;
    for (int i = s; i < e; ++i) {
        int2 cv = csr_cv[i];                      // single b64 broadcast load
        float v = __int_as_float(cv.y);
        const float* x = ego + (long long)cv.x * DIM;
        a0 += v * x[lane];
        a1 += v * x[lane + 32];
    }
    const float* er = ego + (long long)r * DIM;
    h[(long long)r * DIM + lane]      = er[lane] + a0;
    h[(long long)r * DIM + lane + 32] = er[lane + 32] + a1;
}

// ---------------- temp = leaky_relu(h @ fc_w + fc_b), in place, via WMMA f32 ------
// One wave computes a 16-row x 64-col block. A frags per ISA layout:
// lanes 0-15 hold (K=4k, 4k+1) of row M=lane; lanes 16-31 hold (K=4k+2, 4k+3).
// B: VGPR0 holds rows {k0, k0+2}, VGPR1 holds rows {k0+1, k0+3} (row-striped over lanes).
__global__ __launch_bounds__(32) void k_fc_wmma(float* __restrict__ temp,
                                                const float* __restrict__ fc_w,
                                                const float* __restrict__ fc_b, int N) {
    int m0   = blockIdx.x * 16;
    int lane = threadIdx.x;
    int lr   = lane & 15;
    int hi   = lane >> 4;

    int rA = m0 + lr; if (rA > N - 1) rA = N - 1;
    const float* arow = temp + (long long)rA * DIM;

    v2f a[16];
    #pragma unroll
    for (int kc = 0; kc < 16; ++kc) {
        int k0 = kc * 4 + hi * 2;
        a[kc].x = arow[k0];
        a[kc].y = arow[k0 + 1];
    }

    bool fullTile = (m0 + 16 <= N);               // wave-uniform

    #pragma unroll
    for (int nt = 0; nt < 4; ++nt) {
        int n0 = nt * 16;
        v8f c = {0.f, 0.f, 0.f, 0.f, 0.f, 0.f, 0.f, 0.f};
        #pragma unroll
        for (int kc = 0; kc < 16; ++kc) {
            int kb = kc * 4 + hi * 2;
            v2f b;
            b.x = fc_w[(long long)kb * DIM + n0 + lr];
            b.y = fc_w[(long long)(kb + 1) * DIM + n0 + lr];
            c = __builtin_amdgcn_wmma_f32_16x16x4_f32(false, a[kc], false, b,
                                                      (short)0, c, false, false);
        }
        float bias = fc_b[n0 + lr];
        if (fullTile) {                           // hot path: no exec-mask churn
            #pragma unroll
            for (int j = 0; j < 8; ++j) {
                int row = m0 + j + hi * 8;
                float v = c[j] + bias;
                v = (v > 0.f) ? v : 0.01f * v;    // leaky_relu slope 0.01
                temp[(long long)row * DIM + n0 + lr] = v;
            }
        } else {
            #pragma unroll
            for (int j = 0; j < 8; ++j) {
                int row = m0 + j + hi * 8;
                if (row < N) {
                    float v = c[j] + bias;
                    v = (v > 0.f) ? v : 0.01f * v;
                    temp[(long long)row * DIM + n0 + lr] = v;
                }
            }
        }
    }
}

// ---------------- group mask: scores = temp @ fcg_w + fcg_b; top-1 one-hot --------
__global__ void k_group_mask(const float* __restrict__ temp, const float* __restrict__ fcg_w,
                             const float* __restrict__ fcg_b, int* __restrict__ gmask,
                             int nu, int N) {
    int n = blockIdx.x * blockDim.x + threadIdx.x;
    if (n >= N) return;
    float s0 = fcg_b[0], s1 = fcg_b[1], s2 = fcg_b[2], s3 = fcg_b[3];
    const float* t = temp + (long long)n * DIM;
    for (int k = 0; k < DIM; ++k) {
        float v = t[k];
        const float* w = fcg_w + k * 4;
        s0 += v * w[0]; s1 += v * w[1]; s2 += v * w[2]; s3 += v * w[3];
    }
    float m = fmaxf(fmaxf(s0, s1), fmaxf(s2, s3));
    int bits = (s0 == m ? 1 : 0) | (s1 == m ? 2 : 0) | (s2 == m ? 4 : 0) | (s3 == m ? 8 : 0);
    if (n < nu) bits = 0xF;          // users belong to all groups
    gmask[n] = bits;
}

// ---------------- fused 4-group masked SpMM + layer-sum accumulation -------------
__global__ void k_group_spmm(const int* __restrict__ row_off, const int2* __restrict__ csr_cvm,
                             const int* __restrict__ gmask,
                             const float* __restrict__ in0, const float* __restrict__ in1,
                             const float* __restrict__ in2, const float* __restrict__ in3,
                             float* __restrict__ o0, float* __restrict__ o1,
                             float* __restrict__ o2, float* __restrict__ o3,
                             float* __restrict__ total, int N) {
    int gid  = blockIdx.x * blockDim.x + threadIdx.x;
    int r    = gid >> 5;
    int lane = threadIdx.x & 31;
    if (r >= N) return;
    int rm = gmask[r];
    float a00 = 0.f, a01 = 0.f, a10 = 0.f, a11 = 0.f;
    float a20 = 0.f, a21 = 0.f, a30 = 0.f, a31 = 0.f;
    if (rm) {
        int s = row_off[r], e = row_off[r + 1];
        for (int i = s; i < e; ++i) {
            int2 cvm = csr_cvm[i];                // single b64 broadcast load
            int cm   = cvm.x & rm & 0xF;          // edge active iff both endpoints in group
            if (!cm) continue;
            int c    = ((unsigned)cvm.x) >> 4;
            float v  = __int_as_float(cvm.y);
            long long b = (long long)c * DIM;
            if (cm & 1) { a00 += v * in0[b + lane]; a01 += v * in0[b + lane + 32]; }
            if (cm & 2) { a10 += v * in1[b + lane]; a11 += v * in1[b + lane + 32]; }
            if (cm & 4) { a20 += v * in2[b + lane]; a21 += v * in2[b + lane + 32]; }
            if (cm & 8) { a30 += v * in3[b + lane]; a31 += v * in3[b + lane + 32]; }
        }
    }
    long long o = (long long)r * DIM;
    o0[o + lane] = a00; o0[o + lane + 32] = a01;
    o1[o + lane] = a10; o1[o + lane + 32] = a11;
    o2[o + lane] = a20; o2[o + lane + 32] = a21;
    o3[o + lane] = a30; o3[o + lane + 32] = a31;
    total[o + lane]      += 0.2f * (a00 + a10 + a20 + a30);   // LAYER_W
    total[o + lane + 32] += 0.2f * (a01 + a11 + a21 + a31);
}

// ---------------- host -----------------------------------------------------------
extern "C" void kernel_launch(void* const* d_in, const int* in_sizes, int n_in,
                              void* d_out, int out_size, void* d_ws, size_t ws_size,
                              hipStream_t stream) {
    const float* user  = (const float*)d_in[0];
    const float* up    = (const float*)d_in[1];
    const float* fc_w  = (const float*)d_in[2];
    const float* fc_b  = (const float*)d_in[3];
    const float* fcg_w = (const float*)d_in[4];
    const float* fcg_b = (const float*)d_in[5];
    const float* vals  = (const float*)d_in[6];
    const int*   rows  = (const int*)d_in[7];
    const int*   cols  = (const int*)d_in[8];

    int nu = in_sizes[0] / DIM;
    int np = in_sizes[1] / DIM;
    int N  = nu + np;
    int E  = in_sizes[6];
    long long nd = (long long)N * DIM;
    float* out = (float*)d_out;

    // carve workspace
    char* w = (char*)d_ws;
    auto carve = [&](size_t bytes) -> void* {
        void* p = (void*)w;
        w += (bytes + 255) & ~(size_t)255;
        return p;
    };
    float* ego = (float*)carve(nd * 4);
    float* h   = (float*)carve(nd * 4);           // becomes `temp` in place
    float* embA[4]; float* embB[4];
    for (int g = 0; g < 4; ++g) embA[g] = (float*)carve(nd * 4);
    for (int g = 0; g < 4; ++g) embB[g] = (float*)carve(nd * 4);
    int*   row_off = (int*)carve((size_t)(N + 1) * 4);
    int*   cursor  = (int*)carve((size_t)N * 4);
    int*   gmask   = (int*)carve((size_t)N * 4);
    int2*  csr_cv  = (int2*)carve((size_t)E * 8);
    int2*  csr_cvm = (int2*)carve((size_t)E * 8);
    int nScanBlocks = (N + 1 + 255) / 256;
    int*   partials = (int*)carve((size_t)nScanBlocks * 4);
    (void)ws_size; (void)n_in; (void)out_size;

    const int T = 256;
    int ndBlocks   = (int)((nd + T - 1) / T);
    int eBlocks    = (E + T - 1) / T;
    int nBlocks    = (N + T - 1) / T;
    int waveBlocks = (N + 7) / 8;                 // 8 wave32 rows per 256-thread block

    // 1) ego + layer-0 output term
    k_init<<<ndBlocks, T, 0, stream>>>(user, up, ego, out, nu * DIM, nd);
    // 2) CSR build
    k_zero_i32<<<(N + 1 + T - 1) / T, T, 0, stream>>>(row_off, N + 1);
    k_count<<<eBlocks, T, 0, stream>>>(rows, row_off, E);
    k_scan_block<<<nScanBlocks, 256, 0, stream>>>(row_off, N + 1, partials);
    k_scan_partials<<<1, 1024, 0, stream>>>(partials, nScanBlocks);
    k_scan_add<<<nScanBlocks, 256, 0, stream>>>(row_off, N + 1, partials, cursor, N);
    k_scatter<<<eBlocks, T, 0, stream>>>(rows, cols, vals, cursor, csr_cv, E);
    // 3) h = ego + A*ego
    k_spmm_h<<<waveBlocks, T, 0, stream>>>(row_off, csr_cv, ego, h, N);
    // 4) temp = leaky_relu(h @ fc_w + fc_b)   [WMMA f32 16x16x4]
    k_fc_wmma<<<(N + 15) / 16, 32, 0, stream>>>(h, fc_w, fc_b, N);
    // 5) top-1 group routing + edge repack with endpoint masks
    k_group_mask<<<nBlocks, T, 0, stream>>>(h, fcg_w, fcg_b, gmask, nu, N);
    k_pack_mask<<<eBlocks, T, 0, stream>>>(csr_cv, gmask, csr_cvm, E);
    // 6) layers 1..3 of masked propagation (fused over 4 groups)
    k_group_spmm<<<waveBlocks, T, 0, stream>>>(row_off, csr_cvm, gmask,
                                               ego, ego, ego, ego,
                                               embA[0], embA[1], embA[2], embA[3], out, N);
    k_group_spmm<<<waveBlocks, T, 0, stream>>>(row_off, csr_cvm, gmask,
                                               embA[0], embA[1], embA[2], embA[3],
                                               embB[0], embB[1], embB[2], embB[3], out, N);
    k_group_spmm<<<waveBlocks, T, 0, stream>>>(row_off, csr_cvm, gmask,
                                               embB[0], embB[1], embB[2], embB[3],
                                               embA[0], embA[1], embA[2], embA[3], out, N);
}